// CRFLoss_P_67748814127223
// MI455X (gfx1250) — compile-verified
//
#include <hip/hip_runtime.h>
#include <stdint.h>

#define S_LEN     256
#define BATCH     64
#define TAGS      128
#define START_TAG 126
#define END_TAG   127

typedef __attribute__((ext_vector_type(2))) float v2f;
typedef __attribute__((ext_vector_type(8))) float v8f;

// ---------------------------------------------------------------------------
// Async copy of one 64 KiB score tile (128x128 f32) global -> LDS.
// 256 threads * 16 x b128 each = 4096 x 16B = 64 KiB. ASYNCcnt += 16 per wave.
// ---------------------------------------------------------------------------
__device__ __forceinline__ void async_tile_prefetch(const float* gbase,
                                                    const float* lds_tile,
                                                    int tid) {
  uint32_t lbase = (uint32_t)(uintptr_t)lds_tile;   // low 32 bits = LDS byte addr
  uint64_t sbase = (uint64_t)(uintptr_t)gbase;
#pragma unroll
  for (int k = 0; k < 16; ++k) {
    uint32_t off = (uint32_t)((k * 256 + tid) * 16);
    uint32_t ldst = lbase + off;
    asm volatile("global_load_async_to_lds_b128 %0, %1, %2"
                 :: "v"(ldst), "v"(off), "s"(sbase)
                 : "memory");
  }
}

// ---------------------------------------------------------------------------
// Gold-path energy: 16384 gathered elements, one block-partial per block.
// ---------------------------------------------------------------------------
__global__ __launch_bounds__(256) void crf_gold_kernel(
    const float* __restrict__ scores, const int* __restrict__ target,
    const int* __restrict__ mask_gold, float* __restrict__ part_out) {
  __shared__ float red[256];
  const int tid = threadIdx.x;
  const int idx = blockIdx.x * 256 + tid;          // (s*BATCH + b), 0..16383
  const int t = target[idx];
  const float v = scores[(size_t)idx * (TAGS * TAGS) + t] * (float)mask_gold[idx];
  red[tid] = v;
  __syncthreads();
#pragma unroll
  for (int off = 128; off > 0; off >>= 1) {
    if (tid < off) red[tid] += red[tid + off];
    __syncthreads();
  }
  if (tid == 0) part_out[blockIdx.x] = red[0];
}

// ---------------------------------------------------------------------------
// Forward recursion: one block per batch chain. Triple-buffered async tiles.
// 256 threads = 8 waves; wave w owns rows i in [16w,16w+16), lane l owns
// columns j in [4l,4l+4). Cross-wave combine of the 8 per-wave partial sums
// per column is done with V_WMMA_F32_16X16X4_F32 (A = ones).
// ---------------------------------------------------------------------------
__global__ __launch_bounds__(256) void crf_forward_kernel(
    const float* __restrict__ scores, const int* __restrict__ mask_pad,
    float* __restrict__ part_out) {
  __shared__ float tile[3][TAGS * TAGS];   // 3 x 64 KiB stream buffers
  __shared__ float p[TAGS];
  __shared__ float np[TAGS];
  __shared__ float partial[8][TAGS];
  __shared__ float sm;

  const int tid = threadIdx.x;
  const int b   = blockIdx.x;
  const int w   = tid >> 5;
  const int l   = tid & 31;

  // Kick off the stream: stages 1 and 2 in flight before anything else.
  async_tile_prefetch(scores + ((size_t)1 * BATCH + b) * (TAGS * TAGS), tile[1], tid);
  async_tile_prefetch(scores + ((size_t)2 * BATCH + b) * (TAGS * TAGS), tile[2], tid);

  // partition0 = scores[0, b, START_TAG, :]
  if (tid < TAGS)
    p[tid] = scores[(size_t)b * (TAGS * TAGS) + START_TAG * TAGS + tid];

  for (int s = 1; s < S_LEN; ++s) {
    // Retire the oldest stage: in-order async completion, <=2 stages (32 ops)
    // in flight per wave, so cnt<=16 means stage s has landed.
    if (s < S_LEN - 1) asm volatile("s_wait_asynccnt 0x10" ::: "memory");
    else               asm volatile("s_wait_asynccnt 0x0"  ::: "memory");
    __syncthreads();   // tile[s%3] resident for all waves; prev-iter reads done

    if (s + 2 < S_LEN)
      async_tile_prefetch(scores + ((size_t)(s + 2) * BATCH + b) * (TAGS * TAGS),
                          tile[(s + 2) % 3], tid);

    // m = max_i p[i] (wave 0 shuffle reduction, wave32)
    if (tid < 32) {
      float mx = fmaxf(fmaxf(p[tid], p[tid + 32]), fmaxf(p[tid + 64], p[tid + 96]));
#pragma unroll
      for (int d = 16; d > 0; d >>= 1) mx = fmaxf(mx, __shfl_xor(mx, d, 32));
      if (tid == 0) sm = mx;
    }
    __syncthreads();
    const float m = sm;

    // Single-pass rescaled exp-accumulate over this wave's 16 rows.
    const float* tp = tile[s % 3];
    const float* rp = tp + (w * 16) * TAGS + l * 4;
    float ax = 0.f, ay = 0.f, az = 0.f, aw = 0.f;
#pragma unroll
    for (int r = 0; r < 16; ++r) {
      const float  pm = p[w * 16 + r] - m;          // uniform per wave: broadcast
      const float4 sv = *(const float4*)(rp + r * TAGS);
      ax += __expf(sv.x + pm);
      ay += __expf(sv.y + pm);
      az += __expf(sv.z + pm);
      aw += __expf(sv.w + pm);
    }
    *(float4*)&partial[w][l * 4] = make_float4(ax, ay, az, aw);
    __syncthreads();

    // Column-sum the 8 wave partials with WMMA (A = ones 16x4):
    // D[m][n] = sum_k B[k][n]; B lane layout == partial[g][jt+n] directly.
    const int jt = w * 16;
    const int jj = jt + (l & 15);
    const int g0 = (l < 16) ? 0 : 2;
    v8f acc = {};
    v2f A1; A1[0] = 1.0f; A1[1] = 1.0f;
    v2f B1; B1[0] = partial[g0][jj];     B1[1] = partial[g0 + 1][jj];
    acc = __builtin_amdgcn_wmma_f32_16x16x4_f32(false, A1, false, B1,
                                                (short)0, acc, false, false);
    v2f B2; B2[0] = partial[g0 + 4][jj]; B2[1] = partial[g0 + 5][jj];
    acc = __builtin_amdgcn_wmma_f32_16x16x4_f32(false, A1, false, B2,
                                                (short)0, acc, false, false);
    // acc[0] holds the column sum for column jj in every lane.
    if (l < 16) np[jj] = m + __logf(acc[0]);
    __syncthreads();

    const int mk = mask_pad[s * BATCH + b];
    if (mk != 0 && tid < TAGS) p[tid] = np[tid];
    // next-iteration barrier (after s_wait) orders this write vs. reads
  }
  __syncthreads();
  if (tid == 0) part_out[b] = p[END_TAG];
}

// ---------------------------------------------------------------------------
// Deterministic fixed-order final sum: loss = (sum part - sum gold) / BATCH
// ---------------------------------------------------------------------------
__global__ void crf_finalize_kernel(const float* __restrict__ ws,
                                    float* __restrict__ out) {
  float g = 0.f, pt = 0.f;
  for (int i = 0; i < BATCH; ++i) {
    g  += ws[i];
    pt += ws[BATCH + i];
  }
  out[0] = (pt - g) * (1.0f / (float)BATCH);
}

extern "C" void kernel_launch(void* const* d_in, const int* in_sizes, int n_in,
                              void* d_out, int out_size, void* d_ws, size_t ws_size,
                              hipStream_t stream) {
  const float* scores   = (const float*)d_in[0];
  const int*   target   = (const int*)d_in[1];
  const int*   maskgold = (const int*)d_in[2];
  const int*   maskpad  = (const int*)d_in[3];
  float* ws  = (float*)d_ws;    // [0..63] gold partials, [64..127] chain ends
  float* out = (float*)d_out;

  crf_gold_kernel<<<64, 256, 0, stream>>>(scores, target, maskgold, ws);
  crf_forward_kernel<<<BATCH, 256, 0, stream>>>(scores, maskpad, ws + BATCH);
  crf_finalize_kernel<<<1, 1, 0, stream>>>(ws, out);
}